// SVDISTABlock_36661840839102
// MI455X (gfx1250) — compile-verified
//
#include <hip/hip_runtime.h>
#include <hip/hip_bf16.h>
#include <math.h>

// Problem constants (from reference)
#define NB   16          // batch
#define NC   31          // channels (matrix rows)
#define NCP  32          // padded rows
#define HW   65536       // 256*256 columns
#define KCHUNKS 128      // gram k-split
#define KC   (HW / KCHUNKS)       // 512 columns per gram chunk
#define TILES_PER_BLOCK 16        // apply: 16-column tiles per wave
#define NSWEEPS 8

typedef __attribute__((ext_vector_type(2))) float v2f;
typedef __attribute__((ext_vector_type(8))) float v8f;

__device__ inline v8f wmma_f32_k4(v2f a, v2f b, v8f c) {
    // D(16x16,f32) = A(16x4,f32) * B(4x16,f32) + C
    return __builtin_amdgcn_wmma_f32_16x16x4_f32(
        /*neg_a=*/false, a, /*neg_b=*/false, b,
        /*c_mod=*/(short)0, c, /*reuse_a=*/false, /*reuse_b=*/false);
}

// ---------------------------------------------------------------------------
// Phase 0: zero the Gram accumulator region (atomics accumulate into it)
// ---------------------------------------------------------------------------
__global__ void zero_kernel(float* __restrict__ g, int n) {
    int i = blockIdx.x * blockDim.x + threadIdx.x;
    if (i < n) g[i] = 0.0f;
}

// ---------------------------------------------------------------------------
// Phase 1: G[b] = M_b * M_b^T  (32x32, padded), WMMA f32 16x16x4 reduction
// grid = (KCHUNKS, NB), block = 32 (one wave)
// Pad row 31 handled branchlessly: load clamped row, v_cndmask to zero.
// ---------------------------------------------------------------------------
__global__ __launch_bounds__(32) void gram_kernel(const float* __restrict__ x,
                                                  float* __restrict__ G) {
    const int b     = blockIdx.y;
    const int chunk = blockIdx.x;
    const int lane  = threadIdx.x;
    const int half  = lane >> 4;      // 0: K=0,1  1: K=2,3
    const int l     = lane & 15;

    const float* Mb = x + (size_t)b * NC * HW;

    // Per-lane row base pointers (row1 clamped to 30: lane l==15 is the pad
    // row; clamping keeps the load in-bounds even for the last batch).
    const int  row1      = 16 + l;
    const bool row1_ok   = (row1 < NC);
    const int  row1_c    = row1_ok ? row1 : (NC - 1);
    const float* p0 = Mb + (size_t)l      * HW + chunk * KC + 2 * half;
    const float* p1 = Mb + (size_t)row1_c * HW + chunk * KC + 2 * half;

    v8f g00 = {}; v8f g01 = {}; v8f g10 = {}; v8f g11 = {};

    for (int s = 0; s < KC; s += 4) {
        v2f f0 = *(const v2f*)(p0 + s);
        v2f t1 = *(const v2f*)(p1 + s);
        v2f f1;
        f1.x = row1_ok ? t1.x : 0.0f;      // branchless (v_cndmask)
        f1.y = row1_ok ? t1.y : 0.0f;

        // B[k][n] = M[n][k]  ==> B fragment is identical to the A fragment
        g00 = wmma_f32_k4(f0, f0, g00);
        g01 = wmma_f32_k4(f0, f1, g01);
        g10 = wmma_f32_k4(f1, f0, g10);
        g11 = wmma_f32_k4(f1, f1, g11);
    }

    // D layout: VGPR v -> row (v + 8*half), col = l within tile
    float* Gb = G + b * (NCP * NCP);
    const int ro = (half ? 8 : 0);
    #pragma unroll
    for (int v = 0; v < 8; ++v) {
        const int r = v + ro;
        atomicAdd(&Gb[(r)      * NCP + (l)],      g00[v]);
        atomicAdd(&Gb[(r)      * NCP + (16 + l)], g01[v]);
        atomicAdd(&Gb[(16 + r) * NCP + (l)],      g10[v]);
        atomicAdd(&Gb[(16 + r) * NCP + (16 + l)], g11[v]);
    }
}

// ---------------------------------------------------------------------------
// Phase 2: per-batch cyclic Jacobi eigensolve of G (32x32 sym) in LDS,
// then P = V diag(mask) V^T with mask_i = (sqrt(max(lam_i,0)) > thr).
// grid = NB, block = 32 (one wave)
// ---------------------------------------------------------------------------
__global__ __launch_bounds__(32) void jacobi_kernel(const float* __restrict__ G,
                                                    const float* __restrict__ soft_thr,
                                                    float* __restrict__ P) {
    const int b = blockIdx.x;
    const int t = threadIdx.x;

    __shared__ float sG[NCP][NCP + 1];
    __shared__ float sV[NCP][NCP + 1];

    const float* Gb = G + b * (NCP * NCP);
    for (int j = 0; j < NCP; ++j) {
        sG[t][j] = Gb[t * NCP + j];
        sV[t][j] = (t == j) ? 1.0f : 0.0f;
    }
    __syncthreads();

    for (int sweep = 0; sweep < NSWEEPS; ++sweep) {
        for (int p = 0; p < NCP - 1; ++p) {
            for (int q = p + 1; q < NCP; ++q) {
                const float apq = sG[p][q];            // uniform across lanes
                const float app = sG[p][p];
                const float aqq = sG[q][q];
                __syncthreads();
                if (fabsf(apq) > 1e-30f) {             // uniform branch
                    const float tau = (aqq - app) / (2.0f * apq);
                    const float tt  = (tau >= 0.0f ? 1.0f : -1.0f)
                                      / (fabsf(tau) + sqrtf(1.0f + tau * tau));
                    const float cc  = rsqrtf(1.0f + tt * tt);
                    const float ss  = tt * cc;
                    // rows p,q  (G' = J^T G)
                    const float gp = sG[p][t];
                    const float gq = sG[q][t];
                    __syncthreads();
                    sG[p][t] = cc * gp - ss * gq;
                    sG[q][t] = ss * gp + cc * gq;
                    __syncthreads();
                    // cols p,q  (G'' = G' J), and V' = V J
                    const float hp = sG[t][p];
                    const float hq = sG[t][q];
                    const float vp = sV[t][p];
                    const float vq = sV[t][q];
                    __syncthreads();
                    sG[t][p] = cc * hp - ss * hq;
                    sG[t][q] = ss * hp + cc * hq;
                    sV[t][p] = cc * vp - ss * vq;
                    sV[t][q] = ss * vp + cc * vq;
                    __syncthreads();
                }
            }
        }
    }
    __syncthreads();

    const float thr = soft_thr[0];
    float mask[NCP];
    #pragma unroll
    for (int k = 0; k < NCP; ++k) {
        const float lam = sG[k][k];
        const float sv  = sqrtf(fmaxf(lam, 0.0f));
        mask[k] = (sv > thr) ? 1.0f : 0.0f;
    }

    float* Pb = P + b * (NCP * NCP);
    for (int j = 0; j < NCP; ++j) {
        float acc = 0.0f;
        #pragma unroll
        for (int k = 0; k < NCP; ++k)
            acc += mask[k] * sV[t][k] * sV[j][k];
        Pb[t * NCP + j] = acc;
    }
}

// ---------------------------------------------------------------------------
// Phase 3: out[b] = P[b] (32x32) @ M_b (32xHW), streaming WMMA GEMM.
// Non-temporal stores keep x resident in L2 (x: 130 MB < 192 MB L2).
// grid = (HW/16/TILES_PER_BLOCK, NB), block = 32 (one wave)
// ---------------------------------------------------------------------------
__global__ __launch_bounds__(32) void apply_kernel(const float* __restrict__ x,
                                                   const float* __restrict__ P,
                                                   float* __restrict__ out) {
    const int b     = blockIdx.y;
    const int tile0 = blockIdx.x * TILES_PER_BLOCK;
    const int lane  = threadIdx.x;
    const int half  = lane >> 4;
    const int l     = lane & 15;

    const float* Mb = x + (size_t)b * NC * HW;
    const float* Pb = P + b * (NCP * NCP);
    float*       Ob = out + (size_t)b * NC * HW;

    // A fragments of P: row-tile r in {0,1}, k-step s in 0..7
    v2f a[2][8];
    #pragma unroll
    for (int r = 0; r < 2; ++r) {
        #pragma unroll
        for (int s = 0; s < 8; ++s) {
            const int row = 16 * r + l;
            const int col = 4 * s + 2 * half;
            a[r][s] = *(const v2f*)(Pb + row * NCP + col);
        }
    }

    for (int tt = 0; tt < TILES_PER_BLOCK; ++tt) {
        const int j = (tile0 + tt) * 16;
        const int colj = j + l;
        v8f d0 = {}; v8f d1 = {};
        #pragma unroll
        for (int s = 0; s < 8; ++s) {
            const int kk = 4 * s + 2 * half;               // 0..30 (always real)
            const int k2 = kk + 1;                          // 31 only for s=7,half=1
            const bool k2ok = (k2 < NC);
            const int k2c = k2ok ? k2 : (NC - 1);           // clamp: keep load in-bounds
            v2f bf;
            bf.x = Mb[(size_t)kk  * HW + colj];
            {
                const float t = Mb[(size_t)k2c * HW + colj];
                bf.y = k2ok ? t : 0.0f;                     // branchless (v_cndmask)
            }
            d0 = wmma_f32_k4(a[0][s], bf, d0);
            d1 = wmma_f32_k4(a[1][s], bf, d1);
        }
        // D layout: VGPR v -> row (v + 8*half) of the tile, col = l
        const int ro = (half ? 8 : 0);
        #pragma unroll
        for (int v = 0; v < 8; ++v) {
            const int r0 = v + ro;            // rows 0..15
            __builtin_nontemporal_store(d0[v], &Ob[(size_t)r0 * HW + colj]);
            const int r1 = 16 + r0;           // rows 16..31; skip pad row 31
            if (r1 < NC)
                __builtin_nontemporal_store(d1[v], &Ob[(size_t)r1 * HW + colj]);
        }
    }
}

// ---------------------------------------------------------------------------
extern "C" void kernel_launch(void* const* d_in, const int* in_sizes, int n_in,
                              void* d_out, int out_size, void* d_ws, size_t ws_size,
                              hipStream_t stream) {
    const float* x        = (const float*)d_in[0];   // (16,31,256,256) f32
    const float* soft_thr = (const float*)d_in[1];   // (1,) f32

    float* G = (float*)d_ws;                 // 16 * 32*32 floats = 64 KB
    float* P = G + NB * NCP * NCP;           // 16 * 32*32 floats = 64 KB
    float* o = (float*)d_out;

    const int gsz = NB * NCP * NCP;
    zero_kernel<<<(gsz + 255) / 256, 256, 0, stream>>>(G, gsz);

    gram_kernel<<<dim3(KCHUNKS, NB), 32, 0, stream>>>(x, G);

    jacobi_kernel<<<NB, 32, 0, stream>>>(G, soft_thr, P);

    apply_kernel<<<dim3(HW / 16 / TILES_PER_BLOCK, NB), 32, 0, stream>>>(x, P, o);
}